// MvdrBeamformer_78984448573609
// MI455X (gfx1250) — compile-verified
//
#include <hip/hip_runtime.h>
#include <hip/hip_bf16.h>

#define NB   8
#define CH   8
#define FF   257
#define TT   1000
#define ATTD 512
#define NF   (NB * FF)          // 2056
#define EPSILON 1.1920928955078125e-07f
#define EPS5 1e-5f

typedef float v2f __attribute__((ext_vector_type(2)));
typedef float v8f __attribute__((ext_vector_type(8)));

// ---------------------------------------------------------------------------
// Kernel 1a: per-(n,f) mask statistics: max over t, sum over t.
//   mscale  = 1/(max+eps)
//   invden_s = 1/max(sum_m_norm, eps)        (Rs denominator)
//   invden_n = 1/max(T - sum_m_norm, eps)    (Rn denominator)
// mask_s layout: N x T x F (f contiguous -> coalesced across threads)
// ---------------------------------------------------------------------------
__global__ __launch_bounds__(288) void mask_stats_kernel(
    const float* __restrict__ mask, float* __restrict__ mscale,
    float* __restrict__ invden_s, float* __restrict__ invden_n) {
  int n = blockIdx.x;
  int f = threadIdx.x;
  if (f >= FF) return;
  const float* p = mask + (size_t)n * TT * FF + f;
  float mx = 0.0f, sm = 0.0f;
  for (int t = 0; t < TT; ++t) {
    float v = p[(size_t)t * FF];
    mx = fmaxf(mx, fabsf(v));
    sm += v;
  }
  float inv = 1.0f / (mx + EPSILON);
  float msum = sm * inv;
  int nf = n * FF + f;
  mscale[nf]   = inv;
  invden_s[nf] = 1.0f / fmaxf(msum, EPSILON);
  invden_n[nf] = 1.0f / fmaxf((float)TT - msum, EPSILON);
}

// ---------------------------------------------------------------------------
// Kernel 1b: build normalized, transposed mask  mT[n][f][t] (t contiguous)
// via 32x32 LDS tile transpose (coalesced read along f, write along t).
// ---------------------------------------------------------------------------
__global__ __launch_bounds__(256) void mask_transpose_kernel(
    const float* __restrict__ mask, const float* __restrict__ mscale,
    float* __restrict__ mT) {
  __shared__ float tile[32][33];
  int t0 = blockIdx.x * 32;
  int f0 = blockIdx.y * 32;
  int n  = blockIdx.z;
  int tid = threadIdx.x;
  int fl = tid & 31, ts = tid >> 5;   // read: f across lanes
#pragma unroll
  for (int p = 0; p < 4; ++p) {
    int tl = ts + p * 8;
    int f = f0 + fl, t = t0 + tl;
    float v = 0.0f;
    if (f < FF && t < TT) v = mask[((size_t)n * TT + t) * FF + f];
    tile[tl][fl] = v;
  }
  __syncthreads();
  int tl2 = tid & 31, fs = tid >> 5;  // write: t across lanes
#pragma unroll
  for (int p = 0; p < 4; ++p) {
    int fl2 = fs + p * 8;
    int f = f0 + fl2, t = t0 + tl2;
    if (f < FF && t < TT) {
      float sc = mscale[n * FF + f];
      mT[((size_t)n * FF + f) * TT + t] = tile[tl2][fl2] * sc;
    }
  }
}

// ---------------------------------------------------------------------------
// Kernel 2: WMMA covariance. One wave per (n,f).
// X (16 x T) = [re(ch0..7); im(ch0..7)].  Accumulate with V_WMMA_F32_16X16X4:
//   G_m  = X * diag(m) * X^T    (masked)
//   G_t  = X * X^T              (total)  ->  Rn nom = G_t - G_m
// A frag (16x4 f32): lane M = lane&15, K pair = (lane<16 ? 0 : 2) -> float2.
// B frag (4x16 f32): same lane->col mapping -> b = a * m (elementwise).
// Recombine through per-wave LDS 16x16 tiles into Rs/Rn (C x C complex).
// ---------------------------------------------------------------------------
__global__ __launch_bounds__(128) void covar_wmma_kernel(
    const float* __restrict__ xre, const float* __restrict__ xim,
    const float* __restrict__ mT,
    const float* __restrict__ invden_s, const float* __restrict__ invden_n,
    float* __restrict__ Rs_re, float* __restrict__ Rs_im,
    float* __restrict__ Rn_re, float* __restrict__ Rn_im) {
  __shared__ float Lm[4][16][16];
  __shared__ float Lt[4][16][16];
  const int wave = threadIdx.x >> 5;
  const int lane = threadIdx.x & 31;
  const int nf = blockIdx.x * 4 + wave;        // grid*waves == NF exactly
  const int n = nf / FF, f = nf % FF;

  const int row = lane & 15;                   // M (and B column) index
  const int c   = row & 7;
  const int kb  = (lane >> 4) << 1;            // K base: 0 or 2
  const float* xb = (row < 8) ? xre : xim;
  const float* xp = xb + (((size_t)(n * CH + c)) * FF + f) * TT + kb;
  const float* mp = mT + (size_t)nf * TT + kb;

  v8f accm = {};
  v8f acct = {};
#pragma unroll 2
  for (int t0 = 0; t0 < TT; t0 += 4) {
    v2f a  = *(const v2f*)(xp + t0);
    v2f mv = *(const v2f*)(mp + t0);
    v2f b  = a * mv;
    accm = __builtin_amdgcn_wmma_f32_16x16x4_f32(false, a, false, b,
                                                 (short)0, accm, false, false);
    acct = __builtin_amdgcn_wmma_f32_16x16x4_f32(false, a, false, a,
                                                 (short)0, acct, false, false);
  }

  // C/D layout: vgpr r, lanes 0-15 -> (M=r, N=lane); lanes 16-31 -> (M=r+8)
  const int Mb = (lane < 16) ? 0 : 8;
  const int Nn = lane & 15;
#pragma unroll
  for (int r = 0; r < 8; ++r) {
    Lm[wave][Mb + r][Nn] = accm[r];
    Lt[wave][Mb + r][Nn] = acct[r];
  }
  __syncthreads();

  const float invs = invden_s[nf];
  const float invn = invden_n[nf];
#pragma unroll
  for (int q = 0; q < 2; ++q) {
    int e = lane + 32 * q;            // e = i*8 + j
    int i = e >> 3, j = e & 7;
    float gm0 = Lm[wave][i][j],     gm1 = Lm[wave][i + 8][j + 8];
    float gm2 = Lm[wave][i + 8][j], gm3 = Lm[wave][i][j + 8];
    float gt0 = Lt[wave][i][j],     gt1 = Lt[wave][i + 8][j + 8];
    float gt2 = Lt[wave][i + 8][j], gt3 = Lt[wave][i][j + 8];
    size_t o = (size_t)nf * 64 + e;
    Rs_re[o] = (gm0 + gm1) * invs;
    Rs_im[o] = (gm2 - gm3) * invs;
    Rn_re[o] = ((gt0 - gm0) + (gt1 - gm1)) * invn;
    Rn_im[o] = ((gt2 - gm2) - (gt3 - gm3)) * invn;
  }
}

// ---------------------------------------------------------------------------
// Kernel 3: channel-attention feature: feat[n][c][f] = |sum_{j!=c} Rs[c][j]|/7
// ---------------------------------------------------------------------------
__global__ __launch_bounds__(256) void feat_kernel(
    const float* __restrict__ Rs_re, const float* __restrict__ Rs_im,
    float* __restrict__ feat) {
  int idx = blockIdx.x * 256 + threadIdx.x;
  if (idx >= NF * CH) return;
  int nf = idx >> 3, c = idx & 7;
  int n = nf / FF, f = nf % FF;
  float sr = 0.0f, si = 0.0f;
#pragma unroll
  for (int j = 0; j < CH; ++j) {
    if (j != c) {
      sr += Rs_re[(size_t)nf * 64 + c * 8 + j];
      si += Rs_im[(size_t)nf * 64 + c * 8 + j];
    }
  }
  feat[((size_t)(n * CH + c)) * FF + f] = sqrtf(sr * sr + si * si) / 7.0f;
}

// ---------------------------------------------------------------------------
// Kernel 4: g[r] = sum_a tanh(feat_row . proj_w[a] + proj_b[a]) * gvec_w[a] + gvec_b
// One block per row r (= n*C + c).
// ---------------------------------------------------------------------------
__global__ __launch_bounds__(256) void proj_kernel(
    const float* __restrict__ feat, const float* __restrict__ pw,
    const float* __restrict__ pb, const float* __restrict__ gw,
    const float* __restrict__ gb, float* __restrict__ g) {
  __shared__ float frow[FF];
  __shared__ float red[256];
  int r = blockIdx.x, tid = threadIdx.x;
  for (int i = tid; i < FF; i += 256) frow[i] = feat[(size_t)r * FF + i];
  __syncthreads();
  float part = 0.0f;
#pragma unroll
  for (int q = 0; q < ATTD / 256; ++q) {
    int a = tid + q * 256;
    float dot = pb[a];
    const float* wrow = pw + (size_t)a * FF;
    for (int f = 0; f < FF; ++f) dot = fmaf(frow[f], wrow[f], dot);
    part += tanhf(dot) * gw[a];
  }
  red[tid] = part;
  __syncthreads();
  for (int s = 128; s > 0; s >>= 1) {
    if (tid < s) red[tid] += red[tid + s];
    __syncthreads();
  }
  if (tid == 0) g[r] = red[0] + gb[0];
}

// ---------------------------------------------------------------------------
// Kernel 5: softmax over channels per batch (8 values each).
// ---------------------------------------------------------------------------
__global__ void softmax_kernel(const float* __restrict__ g, float* __restrict__ u) {
  int n = threadIdx.x;
  if (n >= NB) return;
  float v[CH], mx = -3.402823466e+38f;
#pragma unroll
  for (int c = 0; c < CH; ++c) { v[c] = g[n * CH + c]; mx = fmaxf(mx, v[c]); }
  float s = 0.0f;
#pragma unroll
  for (int c = 0; c < CH; ++c) { v[c] = expf(v[c] - mx); s += v[c]; }
#pragma unroll
  for (int c = 0; c < CH; ++c) u[n * CH + c] = v[c] / s;
}

// ---------------------------------------------------------------------------
// Kernel 6: per-(n,f) MVDR solve. 8x8 complex Gauss-Jordan inverse of
// Rn + eps*I, then  w = Rn^-1 (Rs u) / (tr(Rn^-1 Rs) + eps).
// One thread per (n,f).
// ---------------------------------------------------------------------------
__global__ __launch_bounds__(64) void mvdr_kernel(
    const float* __restrict__ Rs_re, const float* __restrict__ Rs_im,
    const float* __restrict__ Rn_re, const float* __restrict__ Rn_im,
    const float* __restrict__ u,
    float* __restrict__ w_re, float* __restrict__ w_im) {
  int nf = blockIdx.x * 64 + threadIdx.x;
  if (nf >= NF) return;
  int n = nf / FF;

  float ar[64], ai[64];
  for (int e = 0; e < 64; ++e) {
    ar[e] = Rn_re[(size_t)nf * 64 + e];
    ai[e] = Rn_im[(size_t)nf * 64 + e];
  }
  for (int k = 0; k < 8; ++k) ar[k * 9] += EPS5;

  // in-place complex Gauss-Jordan inversion (no pivoting; diagonal-loaded)
  for (int k = 0; k < 8; ++k) {
    float dr = ar[k * 8 + k], di = ai[k * 8 + k];
    float den = 1.0f / (dr * dr + di * di);
    float idr = dr * den, idi = -di * den;
    ar[k * 8 + k] = idr; ai[k * 8 + k] = idi;
    for (int j = 0; j < 8; ++j) {
      if (j == k) continue;
      float tr = ar[k * 8 + j] * idr - ai[k * 8 + j] * idi;
      float ti = ar[k * 8 + j] * idi + ai[k * 8 + j] * idr;
      ar[k * 8 + j] = tr; ai[k * 8 + j] = ti;
    }
    for (int i = 0; i < 8; ++i) {
      if (i == k) continue;
      float fr = ar[i * 8 + k], fi = ai[i * 8 + k];
      for (int j = 0; j < 8; ++j) {
        if (j == k) continue;
        ar[i * 8 + j] -= fr * ar[k * 8 + j] - fi * ai[k * 8 + j];
        ai[i * 8 + j] -= fr * ai[k * 8 + j] + fi * ar[k * 8 + j];
      }
      ar[i * 8 + k] = -(fr * idr - fi * idi);
      ai[i * 8 + k] = -(fr * idi + fi * idr);
    }
  }

  float uv[8];
#pragma unroll
  for (int c = 0; c < 8; ++c) uv[c] = u[n * 8 + c];

  float yr[8], yi[8];
#pragma unroll
  for (int i = 0; i < 8; ++i) { yr[i] = 0.0f; yi[i] = 0.0f; }
  float trr = 0.0f, tri = 0.0f;
  for (int j = 0; j < 8; ++j) {
    float sr[8], si[8];
    for (int c2 = 0; c2 < 8; ++c2) {
      sr[c2] = Rs_re[(size_t)nf * 64 + j * 8 + c2];
      si[c2] = Rs_im[(size_t)nf * 64 + j * 8 + c2];
      yr[j] += sr[c2] * uv[c2];
      yi[j] += si[c2] * uv[c2];
    }
    for (int i = 0; i < 8; ++i) {   // tr += Rninv[i][j] * Rs[j][i]
      trr += ar[i * 8 + j] * sr[i] - ai[i * 8 + j] * si[i];
      tri += ar[i * 8 + j] * si[i] + ai[i * 8 + j] * sr[i];
    }
  }
  trr += EPS5;
  float tden = 1.0f / (trr * trr + tri * tri);
  float itr = trr * tden, iti = -tri * tden;

  for (int i = 0; i < 8; ++i) {
    float wr0 = 0.0f, wi0 = 0.0f;
    for (int j = 0; j < 8; ++j) {
      wr0 += ar[i * 8 + j] * yr[j] - ai[i * 8 + j] * yi[j];
      wi0 += ar[i * 8 + j] * yi[j] + ai[i * 8 + j] * yr[j];
    }
    w_re[(size_t)nf * 8 + i] = wr0 * itr - wi0 * iti;
    w_im[(size_t)nf * 8 + i] = wr0 * iti + wi0 * itr;
  }
}

// ---------------------------------------------------------------------------
// Kernel 7: beamforming  beam[n][t][f] = sum_c conj(w) * x,  output N,T,F,2.
// Reads coalesced along t; LDS 32x32 tile transpose so writes coalesce on f.
// ---------------------------------------------------------------------------
__global__ __launch_bounds__(256) void beam_kernel(
    const float* __restrict__ xre, const float* __restrict__ xim,
    const float* __restrict__ w_re, const float* __restrict__ w_im,
    float* __restrict__ out) {
  __shared__ float tre[32][33];
  __shared__ float tim[32][33];
  __shared__ float wr[32][8];
  __shared__ float wi[32][8];
  int t0 = blockIdx.x * 32;
  int f0 = blockIdx.y * 32;
  int n  = blockIdx.z;
  int tid = threadIdx.x;

  {
    int fl = tid >> 3, c = tid & 7;
    int f = f0 + fl;
    float a = 0.0f, b = 0.0f;
    if (f < FF) {
      a = w_re[((size_t)n * FF + f) * 8 + c];
      b = w_im[((size_t)n * FF + f) * 8 + c];
    }
    wr[fl][c] = a;
    wi[fl][c] = b;
  }
  __syncthreads();

  int tl = tid & 31, fs = tid >> 5;
  float accr[4] = {0.f, 0.f, 0.f, 0.f};
  float acci[4] = {0.f, 0.f, 0.f, 0.f};
  int t = t0 + tl;
#pragma unroll
  for (int p = 0; p < 4; ++p) {
    int fl = fs + p * 8;
    int f = f0 + fl;
    if (f < FF && t < TT) {
#pragma unroll
      for (int c = 0; c < CH; ++c) {
        size_t o = (((size_t)(n * CH + c)) * FF + f) * TT + t;
        float xr = xre[o], xi = xim[o];
        float a = wr[fl][c], b = wi[fl][c];
        accr[p] += a * xr + b * xi;     // conj(w)*x
        acci[p] += a * xi - b * xr;
      }
    }
  }
#pragma unroll
  for (int p = 0; p < 4; ++p) {
    tre[fs + p * 8][tl] = accr[p];
    tim[fs + p * 8][tl] = acci[p];
  }
  __syncthreads();

  int fl2 = tid & 31, ts = tid >> 5;
#pragma unroll
  for (int p = 0; p < 4; ++p) {
    int tl2 = ts + p * 8;
    int f = f0 + fl2, tt = t0 + tl2;
    if (f < FF && tt < TT) {
      float2 v = make_float2(tre[fl2][tl2], tim[fl2][tl2]);
      *(float2*)(out + (((size_t)n * TT + tt) * FF + f) * 2) = v;
    }
  }
}

// ---------------------------------------------------------------------------
extern "C" void kernel_launch(void* const* d_in, const int* in_sizes, int n_in,
                              void* d_out, int out_size, void* d_ws, size_t ws_size,
                              hipStream_t stream) {
  const float* mask_s = (const float*)d_in[0];
  const float* x_re   = (const float*)d_in[1];
  const float* x_im   = (const float*)d_in[2];
  const float* proj_w = (const float*)d_in[3];
  const float* proj_b = (const float*)d_in[4];
  const float* gvec_w = (const float*)d_in[5];
  const float* gvec_b = (const float*)d_in[6];
  float* out = (float*)d_out;

  float* ws = (float*)d_ws;
  size_t off = 0;
  float* mscale   = ws + off; off += NF;              // 2056
  float* invden_s = ws + off; off += NF;
  float* invden_n = ws + off; off += NF;
  float* mT       = ws + off; off += (size_t)NF * TT; // 2,056,000
  float* Rs_re    = ws + off; off += (size_t)NF * 64;
  float* Rs_im    = ws + off; off += (size_t)NF * 64;
  float* Rn_re    = ws + off; off += (size_t)NF * 64;
  float* Rn_im    = ws + off; off += (size_t)NF * 64;
  float* feat     = ws + off; off += (size_t)NB * CH * FF;
  float* gbuf     = ws + off; off += NB * CH;
  float* ubuf     = ws + off; off += NB * CH;
  float* w_re     = ws + off; off += (size_t)NF * CH;
  float* w_im     = ws + off; off += (size_t)NF * CH;

  // 1a: mask stats
  mask_stats_kernel<<<NB, 288, 0, stream>>>(mask_s, mscale, invden_s, invden_n);
  // 1b: normalized transposed mask
  mask_transpose_kernel<<<dim3((TT + 31) / 32, (FF + 31) / 32, NB), 256, 0, stream>>>(
      mask_s, mscale, mT);
  // 2: WMMA covariances (4 waves/block, 1 wave per (n,f); NF = 514*4)
  covar_wmma_kernel<<<NF / 4, 128, 0, stream>>>(
      x_re, x_im, mT, invden_s, invden_n, Rs_re, Rs_im, Rn_re, Rn_im);
  // 3: channel-attention features
  feat_kernel<<<(NF * CH + 255) / 256, 256, 0, stream>>>(Rs_re, Rs_im, feat);
  // 4: projection + tanh + gvec dot
  proj_kernel<<<NB * CH, 256, 0, stream>>>(feat, proj_w, proj_b, gvec_w, gvec_b, gbuf);
  // 5: softmax over channels
  softmax_kernel<<<1, 32, 0, stream>>>(gbuf, ubuf);
  // 6: MVDR solve
  mvdr_kernel<<<(NF + 63) / 64, 64, 0, stream>>>(
      Rs_re, Rs_im, Rn_re, Rn_im, ubuf, w_re, w_im);
  // 7: beamform
  beam_kernel<<<dim3((TT + 31) / 32, (FF + 31) / 32, NB), 256, 0, stream>>>(
      x_re, x_im, w_re, w_im, out);
}